// GNN_19585050869931
// MI455X (gfx1250) — compile-verified
//
#include <hip/hip_runtime.h>

typedef float v2f __attribute__((ext_vector_type(2)));
typedef float v8f __attribute__((ext_vector_type(8)));

// ---------------- degree / dinv ----------------
__global__ void k_deg_init(float* deg, int n) {
    int i = blockIdx.x * blockDim.x + threadIdx.x;
    if (i < n) deg[i] = 1.0f;                 // self-loop contributes 1 to every degree
}

__global__ void k_deg_count(const int* __restrict__ dst, float* deg, int E) {
    int e = blockIdx.x * blockDim.x + threadIdx.x;
    if (e < E) atomicAdd(&deg[dst[e]], 1.0f);
}

__global__ void k_dinv(float* deg, int n) {
    int i = blockIdx.x * blockDim.x + threadIdx.x;
    if (i < n) deg[i] = rsqrtf(deg[i]);       // deg >= 1 always (self-loops)
}

// ---------------- fp32 WMMA GEMM: H[M,NOUT] = A[M,K] @ W[K,NOUT] ----------------
// One wave per 16x16 output tile; V_WMMA_F32_16X16X4_F32, K stepped by 4.
// A frag (16x4 f32, 2 VGPR): lane L holds A[L&15][2*(L>>4)+j], j=0,1
// B frag (4x16 f32, 2 VGPR): lane L holds B[2*(L>>4)+j][L&15]
// C/D  (16x16 f32, 8 VGPR):  lane L reg r holds C[r+8*(L>>4)][L&15]
template <int K, int NOUT>
__global__ void k_gemm_wmma(const float* __restrict__ A, const float* __restrict__ W,
                            float* __restrict__ H, int mtiles) {
    int wid  = (blockIdx.x * blockDim.x + threadIdx.x) >> 5;
    int lane = threadIdx.x & 31;
    const int ntiles = NOUT >> 4;
    int tm = wid / ntiles;
    int tn = wid - tm * ntiles;
    if (tm >= mtiles) return;                 // wave-uniform exit: EXEC stays all-1s for WMMA

    int half = lane >> 4;                     // 0: lanes 0-15, 1: lanes 16-31
    int l15  = lane & 15;
    int row  = (tm << 4) + l15;
    const float* arow = A + row * K;
    const int ncol = (tn << 4) + l15;

    v8f acc = {0.f, 0.f, 0.f, 0.f, 0.f, 0.f, 0.f, 0.f};
#pragma unroll
    for (int k = 0; k < K; k += 4) {
        int ka = k + (half << 1);
        v2f av;
        av.x = arow[ka];
        av.y = arow[ka + 1];
        v2f bv;
        bv.x = W[ka * NOUT + ncol];
        bv.y = W[(ka + 1) * NOUT + ncol];
        acc = __builtin_amdgcn_wmma_f32_16x16x4_f32(
            /*neg_a=*/false, av, /*neg_b=*/false, bv,
            /*c_mod=*/(short)0, acc, /*reuse_a=*/false, /*reuse_b=*/false);
    }
#pragma unroll
    for (int r = 0; r < 8; ++r) {
        H[((tm << 4) + r + (half << 3)) * NOUT + ncol] = acc[r];
    }
}

// ---------------- tiny layer-3 GEMM: [N,32] @ [32,2] ----------------
__global__ void k_gemm32x2(const float* __restrict__ A, const float* __restrict__ W,
                           float* __restrict__ H, int n) {
    int i = blockIdx.x * blockDim.x + threadIdx.x;
    if (i >= n) return;
    float o0 = 0.f, o1 = 0.f;
#pragma unroll
    for (int k = 0; k < 32; ++k) {
        float a = A[i * 32 + k];
        o0 += a * W[k * 2 + 0];
        o1 += a * W[k * 2 + 1];
    }
    H[i * 2 + 0] = o0;
    H[i * 2 + 1] = o1;
}

// ---------------- aggregation: init with self-loop term dinv[i]^2 * h[i] ----------------
template <int F>
__global__ void k_selfloop(const float* __restrict__ h, const float* __restrict__ dinv,
                           float* __restrict__ agg, int n) {
    int idx = blockIdx.x * blockDim.x + threadIdx.x;
    if (idx >= n * F) return;
    int i = idx / F;
    float w = dinv[i];
    agg[idx] = w * w * h[idx];
}

// ---------------- edge scatter: agg[dst] += dinv[src]*dinv[dst] * h[src] ----------------
// one thread per (edge, 4-feature chunk); float4 gather (L2-resident) + 4 fp32 atomics
template <int F, int CSHIFT>  // F/4 == 1<<CSHIFT
__global__ void k_scatter(const int* __restrict__ src, const int* __restrict__ dst,
                          const float* __restrict__ dinv, const float* __restrict__ h,
                          float* __restrict__ agg, int E) {
    int idx = blockIdx.x * blockDim.x + threadIdx.x;
    if (idx >= (E << CSHIFT)) return;
    int e = idx >> CSHIFT;
    int c = idx & ((1 << CSHIFT) - 1);
    int s = src[e], d = dst[e];
    float w = dinv[s] * dinv[d];
    const float4* hp = (const float4*)(h + s * F);
    float4 v = hp[c];
    float* out = agg + d * F + (c << 2);
    atomicAdd(out + 0, w * v.x);
    atomicAdd(out + 1, w * v.y);
    atomicAdd(out + 2, w * v.z);
    atomicAdd(out + 3, w * v.w);
}

__global__ void k_scatter2(const int* __restrict__ src, const int* __restrict__ dst,
                           const float* __restrict__ dinv, const float* __restrict__ h,
                           float* __restrict__ agg, int E) {
    int e = blockIdx.x * blockDim.x + threadIdx.x;
    if (e >= E) return;
    int s = src[e], d = dst[e];
    float w = dinv[s] * dinv[d];
    atomicAdd(&agg[d * 2 + 0], w * h[s * 2 + 0]);
    atomicAdd(&agg[d * 2 + 1], w * h[s * 2 + 1]);
}

// ---------------- bias + optional ReLU ----------------
template <int F, bool RELU>
__global__ void k_bias_act(const float* __restrict__ agg, const float* __restrict__ b,
                           float* __restrict__ out, int n) {
    int idx = blockIdx.x * blockDim.x + threadIdx.x;
    if (idx >= n * F) return;
    float v = agg[idx] + b[idx & (F - 1)];
    out[idx] = RELU ? fmaxf(v, 0.f) : v;
}

static inline int cdiv(int a, int b) { return (a + b - 1) / b; }

extern "C" void kernel_launch(void* const* d_in, const int* in_sizes, int n_in,
                              void* d_out, int out_size, void* d_ws, size_t ws_size,
                              hipStream_t stream) {
    (void)n_in; (void)out_size; (void)ws_size;

    const float* x  = (const float*)d_in[0];
    const int*   ei = (const int*)d_in[1];
    const float* W1 = (const float*)d_in[2];
    const float* b1 = (const float*)d_in[3];
    const float* W2 = (const float*)d_in[4];
    const float* b2 = (const float*)d_in[5];
    const float* W3 = (const float*)d_in[6];
    const float* b3 = (const float*)d_in[7];

    const int N = in_sizes[0] / 64;     // 100000 (divisible by 16)
    const int E = in_sizes[1] / 2;      // 1600000
    const int* src = ei;                // edge_index[0]
    const int* dst = ei + E;            // edge_index[1]

    float* ws   = (float*)d_ws;
    float* dinv = ws;                   // N floats (deg -> dinv in place)
    float* bufA = ws + N;               // N*64 floats
    float* bufB = bufA + N * 64;        // N*64 floats

    const int BT = 256;                 // 8 waves per block (wave32)

    // degrees -> dinv
    k_deg_init <<<cdiv(N, BT), BT, 0, stream>>>(dinv, N);
    k_deg_count<<<cdiv(E, BT), BT, 0, stream>>>(dst, dinv, E);
    k_dinv     <<<cdiv(N, BT), BT, 0, stream>>>(dinv, N);

    const int mtiles = N / 16;          // 6250

    // ---- Layer 1: x[N,64] @ W1[64,64] -> relu(agg + b1) ----
    {
        int waves = mtiles * (64 / 16);
        k_gemm_wmma<64, 64><<<cdiv(waves * 32, BT), BT, 0, stream>>>(x, W1, bufA, mtiles);
        k_selfloop<64>     <<<cdiv(N * 64, BT), BT, 0, stream>>>(bufA, dinv, bufB, N);
        k_scatter<64, 4>   <<<cdiv(E * 16, BT), BT, 0, stream>>>(src, dst, dinv, bufA, bufB, E);
        k_bias_act<64,true><<<cdiv(N * 64, BT), BT, 0, stream>>>(bufB, b1, bufA, N);
    }
    // ---- Layer 2: a1[N,64] @ W2[64,32] -> relu(agg + b2) ----
    {
        int waves = mtiles * (32 / 16);
        k_gemm_wmma<64, 32><<<cdiv(waves * 32, BT), BT, 0, stream>>>(bufA, W2, bufB, mtiles);
        k_selfloop<32>     <<<cdiv(N * 32, BT), BT, 0, stream>>>(bufB, dinv, bufA, N);
        k_scatter<32, 3>   <<<cdiv(E * 8, BT), BT, 0, stream>>>(src, dst, dinv, bufB, bufA, E);
        k_bias_act<32,true><<<cdiv(N * 32, BT), BT, 0, stream>>>(bufA, b2, bufB, N);
    }
    // ---- Layer 3: a2[N,32] @ W3[32,2] -> agg + b3 ----
    {
        k_gemm32x2         <<<cdiv(N, BT), BT, 0, stream>>>(bufB, W3, bufA, N);
        k_selfloop<2>      <<<cdiv(N * 2, BT), BT, 0, stream>>>(bufA, dinv, bufB, N);
        k_scatter2         <<<cdiv(E, BT), BT, 0, stream>>>(src, dst, dinv, bufA, bufB, E);
        k_bias_act<2,false><<<cdiv(N * 2, BT), BT, 0, stream>>>(bufB, b3, (float*)d_out, N);
    }
}